// SoftPatternClassifier_41111426957519
// MI455X (gfx1250) — compile-verified
//
#include <hip/hip_runtime.h>
#include <hip/hip_bf16.h>
#include <math.h>

// Problem constants (match reference)
#define B_DIM   64
#define L_DIM   512
#define V_DIM   20000
#define P_DIM   150
#define M_DIM   6
#define K_DIM   300
#define PM_DIM  900            // P*M
#define TM_STRIDE 750          // 5*P : table stored as [v][m][p], m=0..4 only
#define NEG     (-1.0e9f)
#define NT_GROUPS 313          // ceil(1250 n-tiles / 4 per wave)
#define PF_DIST 16             // prefetch distance (tokens) in scan

typedef float v2f __attribute__((ext_vector_type(2)));
typedef float v8f __attribute__((ext_vector_type(8)));

__device__ __forceinline__ float log_sigmoidf(float x) {
    // -log(1 + e^{-x}), numerically stable
    float z = fminf(x, 0.0f);
    return z - log1pf(expf(-fabsf(x)));
}

// -----------------------------------------------------------------------------
// Kernel 1: C(900,20000) = diags @ emb  via V_WMMA_F32_16X16X4_F32 (f32, K=4),
// fused epilogue: +bias, log_sigmoid, max with log_sigmoid(wildcard), store
// transposed+re-laid-out table tmw[v][m][p] (m<5 only). 60 MB -> L2 resident.
// One wave per block; each wave owns a 16x64 output strip.
// -----------------------------------------------------------------------------
__global__ __launch_bounds__(32)
void spp_gemm_wmma(const float* __restrict__ diags,     // (900,300)
                   const float* __restrict__ emb,       // (300,20000)
                   const float* __restrict__ bias,      // (900,)
                   const float* __restrict__ wildcards, // (150,5)
                   float* __restrict__ tmw)             // (20000,750)
{
    const int mt   = blockIdx.x / NT_GROUPS;   // 0..56  (M tile)
    const int ng   = blockIdx.x % NT_GROUPS;   // 0..312 (group of 4 N tiles)
    const int lane = threadIdx.x;
    const int lo   = lane & 15;                // N / M-row selector
    const int hi   = lane >> 4;                // K half selector

    // A fragment source: row = mt*16 + lo, clamp rows >= 900 (guarded at store)
    int arow = mt * 16 + lo;
    if (arow >= PM_DIM) arow = PM_DIM - 1;
    const float* aptr = diags + (size_t)arow * K_DIM + 2 * hi;

    // B columns for the 4 tiles (clamped; invalid tiles skipped at store)
    const int vbase = ng * 64;
    int col[4];
    bool tile_ok[4];
#pragma unroll
    for (int j = 0; j < 4; ++j) {
        int c = vbase + j * 16 + lo;
        tile_ok[j] = (vbase + j * 16 + 16) <= V_DIM;  // V%16==0 -> uniform per tile
        col[j] = c < V_DIM ? c : (V_DIM - 1);
    }

    v8f acc[4] = {v8f{}, v8f{}, v8f{}, v8f{}};

    for (int k = 0; k < K_DIM; k += 4) {
        // A frag: lane holds A[row][k + 2*hi + {0,1}]  (8B-aligned float2)
        v2f a = *(const v2f*)(aptr + k);
        const int r0 = k + 2 * hi;
        const float* e0 = emb + (size_t)r0 * V_DIM;
        const float* e1 = e0 + V_DIM;
#pragma unroll
        for (int j = 0; j < 4; ++j) {
            v2f bb;
            bb.x = e0[col[j]];
            bb.y = e1[col[j]];
            acc[j] = __builtin_amdgcn_wmma_f32_16x16x4_f32(
                /*neg_a=*/false, a, /*neg_b=*/false, bb,
                /*c_mod=*/(short)0, acc[j],
                /*reuse_a=*/false, /*reuse_b=*/false);
        }
    }

    // Epilogue: lane holds D rows pm = mt*16 + 8*hi + r, col v = vbase+j*16+lo
    const int pm0 = mt * 16 + 8 * hi;
#pragma unroll
    for (int j = 0; j < 4; ++j) {
        if (!tile_ok[j]) continue;
        const int v = vbase + j * 16 + lo;
#pragma unroll
        for (int r = 0; r < 8; ++r) {
            const int pm = pm0 + r;
            if (pm >= PM_DIM) continue;
            const int p = pm / M_DIM;
            const int m = pm - p * M_DIM;
            if (m == 5) continue;                       // state-5 column never read
            float x  = acc[j][r] + bias[pm];
            float ls = log_sigmoidf(x);
            float wl = log_sigmoidf(wildcards[p * 5 + m]);
            tmw[(size_t)v * TM_STRIDE + m * P_DIM + p] = fmaxf(ls, wl);
        }
    }
}

// -----------------------------------------------------------------------------
// Kernel 2: per-batch max-plus scan over L=512 tokens + layernorm/heaviside/
// linear epilogue. One block per batch, lane = pattern p (150 active of 160).
// Table reads are coalesced (stride-4B across lanes) and L2-resident.
// Latency hiding (scan is latency-bound, ~320 waves on the whole chip):
//   - token ids staged in LDS once per block
//   - global_prefetch of tmw rows PF_DIST tokens ahead (L2 -> WGP$)
//   - depth-2 register pipeline (load t+2 while computing t)
// -----------------------------------------------------------------------------
__global__ __launch_bounds__(160)
void spp_scan(const int* __restrict__ docs,      // (64,512)
              const int* __restrict__ doc_lens,  // (64,)
              const float* __restrict__ tmw,     // (20000,750)
              const float* __restrict__ ln_gamma,
              const float* __restrict__ ln_beta,
              const float* __restrict__ lw,      // (2,150)
              const float* __restrict__ lb,      // (2,)
              float* __restrict__ out)           // (64,2)
{
    const int b = blockIdx.x;
    const int p = threadIdx.x;

    __shared__ int   sdocs[L_DIM];
    __shared__ float sh[160];
    __shared__ float stats[2];

    // Stage the block's token stream into LDS (2 KB)
    for (int i = p; i < L_DIM; i += 160) sdocs[i] = docs[b * L_DIM + i];
    __syncthreads();

    float sc = NEG;
    if (p < P_DIM) {
        const int e  = 3 + p / 50;            // end state for this pattern
        const int dl = doc_lens[b];

        float h1 = NEG, h2 = NEG, h3 = NEG, h4 = NEG, h5 = NEG;

        // Warm the cache for the first PF_DIST tokens
#pragma unroll
        for (int t = 0; t < PF_DIST; ++t) {
            const float* pr = tmw + (size_t)sdocs[t] * TM_STRIDE + p;
#pragma unroll
            for (int m = 0; m < 5; ++m) __builtin_prefetch(pr + m * P_DIM, 0, 0);
        }

        // Depth-2 register pipeline: cur = values for t, nxt = values for t+1
        const float* b0 = tmw + (size_t)sdocs[0] * TM_STRIDE + p;
        float c0 = b0[0 * P_DIM], c1 = b0[1 * P_DIM], c2 = b0[2 * P_DIM],
              c3 = b0[3 * P_DIM], c4 = b0[4 * P_DIM];
        const float* b1 = tmw + (size_t)sdocs[1] * TM_STRIDE + p;
        float x0 = b1[0 * P_DIM], x1 = b1[1 * P_DIM], x2 = b1[2 * P_DIM],
              x3 = b1[3 * P_DIM], x4 = b1[4 * P_DIM];

        for (int t = 0; t < L_DIM; ++t) {
            // Prefetch the row PF_DIST tokens ahead into near cache
            if (t + PF_DIST < L_DIM) {
                const float* pr = tmw + (size_t)sdocs[t + PF_DIST] * TM_STRIDE + p;
#pragma unroll
                for (int m = 0; m < 5; ++m) __builtin_prefetch(pr + m * P_DIM, 0, 0);
            }
            // Issue loads for t+2 (lands two iterations from now)
            float f0 = 0.f, f1 = 0.f, f2 = 0.f, f3 = 0.f, f4 = 0.f;
            if (t + 2 < L_DIM) {
                const float* nb = tmw + (size_t)sdocs[t + 2] * TM_STRIDE + p;
                f0 = nb[0 * P_DIM]; f1 = nb[1 * P_DIM]; f2 = nb[2 * P_DIM];
                f3 = nb[3 * P_DIM]; f4 = nb[4 * P_DIM];
            }
            // hid_new[m] = hid[m-1] + max(tm,wild)[m-1]; hid[0] == 0 always
            h5 = h4 + c4;
            h4 = h3 + c3;
            h3 = h2 + c2;
            h2 = h1 + c1;
            h1 = c0;
            float esv = (e == 3) ? h3 : ((e == 4) ? h4 : h5);
            if (t < dl) sc = fmaxf(sc, esv);
            // rotate pipeline
            c0 = x0; c1 = x1; c2 = x2; c3 = x3; c4 = x4;
            x0 = f0; x1 = f1; x2 = f2; x3 = f3; x4 = f4;
        }
    }

    float sexp = (p < P_DIM) ? expf(sc) : 0.0f;
    sh[p] = sexp;
    __syncthreads();

    if (p == 0) {
        float s = 0.0f;
        for (int i = 0; i < P_DIM; ++i) s += sh[i];
        const float mu = s / (float)P_DIM;
        float ss = 0.0f;
        for (int i = 0; i < P_DIM; ++i) {
            float d = sh[i] - mu;
            ss += d * d;
        }
        stats[0] = mu;
        stats[1] = ss / (float)P_DIM;
    }
    __syncthreads();

    float co0 = 0.0f, co1 = 0.0f;
    if (p < P_DIM) {
        const float mu = stats[0], var = stats[1];
        float norm = (sexp - mu) * rsqrtf(var + 1e-5f) * ln_gamma[p] + ln_beta[p];
        float bin  = norm > 0.0f ? 1.0f : 0.0f;
        co0 = bin * lw[p];
        co1 = bin * lw[P_DIM + p];
    }

    sh[p] = co0;
    __syncthreads();
    if (p == 0) {
        float s = 0.0f;
        for (int i = 0; i < P_DIM; ++i) s += sh[i];
        out[b * 2 + 0] = s + lb[0];
    }
    __syncthreads();
    sh[p] = co1;
    __syncthreads();
    if (p == 0) {
        float s = 0.0f;
        for (int i = 0; i < P_DIM; ++i) s += sh[i];
        out[b * 2 + 1] = s + lb[1];
    }
}

// -----------------------------------------------------------------------------
extern "C" void kernel_launch(void* const* d_in, const int* in_sizes, int n_in,
                              void* d_out, int out_size, void* d_ws, size_t ws_size,
                              hipStream_t stream) {
    const int*   docs      = (const int*)d_in[0];
    const int*   doc_lens  = (const int*)d_in[1];
    const float* emb       = (const float*)d_in[2];   // (300,20000)
    const float* diags     = (const float*)d_in[3];   // (900,300)
    const float* bias      = (const float*)d_in[4];   // (900,1)
    const float* wildcards = (const float*)d_in[5];   // (150,5)
    const float* ln_gamma  = (const float*)d_in[6];
    const float* ln_beta   = (const float*)d_in[7];
    const float* lw        = (const float*)d_in[8];   // (2,150)
    const float* lb        = (const float*)d_in[9];
    float*       out       = (float*)d_out;           // (64,2)
    float*       tmw       = (float*)d_ws;            // 20000*750 floats = 60 MB

    (void)in_sizes; (void)n_in; (void)out_size; (void)ws_size;

    spp_gemm_wmma<<<57 * NT_GROUPS, 32, 0, stream>>>(diags, emb, bias, wildcards, tmw);
    spp_scan<<<B_DIM, 160, 0, stream>>>(docs, doc_lens, tmw, ln_gamma, ln_beta, lw, lb, out);
}